// Aggregator_36842229465535
// MI455X (gfx1250) — compile-verified
//
#include <hip/hip_runtime.h>
#include <hip/hip_bf16.h>

#define D_DIM 128
#define RRELU_SLOPE 0.22916666666666666f   // (1/8 + 1/3)/2

typedef __attribute__((ext_vector_type(2))) float v2f;
typedef __attribute__((ext_vector_type(8))) float v8f;

// ---------------------------------------------------------------------------
// Kernel 1: gather h0[n, :] = ent_embeds[node_id[n], :]
// ---------------------------------------------------------------------------
__global__ __launch_bounds__(128) void rgcn_gather(
    const float* __restrict__ ent, const int* __restrict__ node_id,
    float* __restrict__ h, int n_nodes) {
  const int node = blockIdx.x;
  if (node >= n_nodes) return;
  const int d = threadIdx.x;
  const int id = node_id[node];           // uniform -> scalar load
  h[(size_t)node * D_DIM + d] = ent[(size_t)id * D_DIM + d];
}

// ---------------------------------------------------------------------------
// Kernel 2: per-edge block-diagonal message + scatter-add into agg.
// One block (128 threads) per edge; thread t computes output element t:
//   b = t/2, o = t%2, msg = h[src][2b]*w[b,0,o] + h[src][2b+1]*w[b,1,o]
// ---------------------------------------------------------------------------
__device__ __forceinline__ void atomic_add_f32(float* p, float v) {
  (void)__hip_atomic_fetch_add(p, v, __ATOMIC_RELAXED, __HIP_MEMORY_SCOPE_AGENT);
}

__global__ __launch_bounds__(128) void rgcn_edge_msg(
    const float* __restrict__ h,          // [N,128]
    const float* __restrict__ weight_l,   // [R,256]  (R,B=64,SI=2,SO=2)
    const int* __restrict__ edge_type,
    const int* __restrict__ src,
    const int* __restrict__ dst,
    float* __restrict__ agg,              // [N,128]
    int n_edges) {
  const int e = blockIdx.x;
  if (e >= n_edges) return;
  const int t  = threadIdx.x;             // 0..127
  const int et = edge_type[e];            // uniform
  const int s  = src[e];                  // uniform
  const int d  = dst[e];                  // uniform

  const float* __restrict__ wrow = weight_l + (size_t)et * 256;
  __builtin_prefetch(wrow, 0, 1);         // global_prefetch_b8: weight row is L2-hot

  const int b = t >> 1;
  const int o = t & 1;
  const v2f hn = *(const v2f*)(h + (size_t)s * D_DIM + 2 * b);
  const float w0 = wrow[4 * b + o];       // (b, i=0, o)
  const float w1 = wrow[4 * b + 2 + o];   // (b, i=1, o)
  const float msg = hn.x * w0 + hn.y * w1;
  atomic_add_f32(agg + (size_t)d * D_DIM + t, msg);
}

// ---------------------------------------------------------------------------
// Kernel 3: out = leaky( agg*norm + h @ W ), via V_WMMA_F32_16X16X4_F32.
// Block: 256 threads = 8 waves. Block owns 16 output rows; wave w owns the
// 16-column tile [16w, 16w+16). W (64 KB) is staged in LDS.
// In-place (out == h) is safe: each block reads only its own 16 rows of h,
// and a barrier separates all A-reads from the epilogue writes.
//
// ISA layouts (wave32):
//   A 16x4 f32 : lane -> M = lane%16 ; a[j] = A[M][2*(lane/16)+j]
//   B 4x16 f32 : lane -> N = lane%16 ; b[j] = B[2*(lane/16)+j][N]
//   C/D 16x16  : vgpr r -> M = r + 8*(lane/16), N = lane%16
// ---------------------------------------------------------------------------
__global__ __launch_bounds__(256) void rgcn_selfloop_gemm(
    const float* __restrict__ h,      // [N,128] (A)
    const float* __restrict__ W,      // [128,128] row-major (k, n)
    const float* __restrict__ agg,    // [N,128]
    const float* __restrict__ norm,   // [N]
    float* __restrict__ out,          // [N,128] (may alias h)
    int n_rows) {
  __shared__ float wlds[D_DIM * D_DIM];   // 64 KB of 320 KB WGP LDS
  for (int i = threadIdx.x; i < D_DIM * D_DIM; i += 256) wlds[i] = W[i];
  __syncthreads();

  const int wave = threadIdx.x >> 5;      // 0..7  -> column tile
  const int lane = threadIdx.x & 31;
  const int half = lane >> 4;             // 0|1
  const int l16  = lane & 15;
  const int row0 = blockIdx.x * 16;
  const int col0 = wave * 16;

  const int m_a   = row0 + l16;
  const int m_ld  = m_a < n_rows ? m_a : (n_rows - 1);  // clamp loads (EXEC all-1s)
  const int kgrp  = half * 2;

  v8f c = {0.f, 0.f, 0.f, 0.f, 0.f, 0.f, 0.f, 0.f};

  for (int k = 0; k < D_DIM; k += 4) {
    v2f a = *(const v2f*)(h + (size_t)m_ld * D_DIM + k + kgrp);
    v2f b;
    b.x = wlds[(k + kgrp + 0) * D_DIM + col0 + l16];
    b.y = wlds[(k + kgrp + 1) * D_DIM + col0 + l16];
    // (neg_a, A, neg_b, B, c_mod, C, reuse_a, reuse_b)
    c = __builtin_amdgcn_wmma_f32_16x16x4_f32(false, a, false, b,
                                              (short)0, c, false, false);
  }

  __syncthreads();  // all A-reads of this block's rows complete before writes

  const int n = col0 + l16;
  const int mbase = row0 + half * 8;
#pragma unroll
  for (int r = 0; r < 8; ++r) {
    const int m = mbase + r;
    if (m < n_rows) {
      float v = c[r] + agg[(size_t)m * D_DIM + n] * norm[m];
      out[(size_t)m * D_DIM + n] = v >= 0.f ? v : v * RRELU_SLOPE;
    }
  }
}

// ---------------------------------------------------------------------------
// Launch: gather -> (zero agg -> edge scatter -> wmma gemm+epilogue) x2
// ---------------------------------------------------------------------------
extern "C" void kernel_launch(void* const* d_in, const int* in_sizes, int n_in,
                              void* d_out, int out_size, void* d_ws, size_t ws_size,
                              hipStream_t stream) {
  const float* ent       = (const float*)d_in[0];
  // d_in[1] = rel_embeds (unused by the layers)
  const float* weight    = (const float*)d_in[2];  // [L, R, 256]
  const float* loopw     = (const float*)d_in[3];  // [L, 128, 128]
  const float* norm      = (const float*)d_in[4];  // [N]
  const int*   node_id   = (const int*)d_in[5];    // [N]
  const int*   edge_type = (const int*)d_in[6];    // [E]
  const int*   src       = (const int*)d_in[7];    // [E]
  const int*   dst       = (const int*)d_in[8];    // [E]

  const int N = in_sizes[5];
  const int E = in_sizes[6];
  const int R = in_sizes[2] / (2 * 256);           // L=2, B*SI*SO=256
  (void)R; (void)n_in; (void)ws_size;

  float* out = (float*)d_out;
  float* h0  = (float*)d_ws;                       // N*128 floats
  float* agg = h0 + (size_t)N * D_DIM;             // N*128 floats

  const size_t aggBytes = (size_t)N * D_DIM * sizeof(float);
  const int gemmGrid = (N + 15) / 16;

  // h0 = ent_embeds[node_id]
  rgcn_gather<<<N, D_DIM, 0, stream>>>(ent, node_id, h0, N);

  // ---- layer 0: h0 -> out ----
  hipMemsetAsync(agg, 0, aggBytes, stream);
  rgcn_edge_msg<<<E, D_DIM, 0, stream>>>(h0, weight, edge_type, src, dst, agg, E);
  rgcn_selfloop_gemm<<<gemmGrid, 256, 0, stream>>>(h0, loopw, agg, norm, out, N);

  // ---- layer 1: out -> out (in-place GEMM is block-row safe) ----
  hipMemsetAsync(agg, 0, aggBytes, stream);
  rgcn_edge_msg<<<E, D_DIM, 0, stream>>>(out, weight + (size_t)R * 256,
                                         edge_type, src, dst, agg, E);
  rgcn_selfloop_gemm<<<gemmGrid, 256, 0, stream>>>(out, loopw + D_DIM * D_DIM,
                                                   agg, norm, out, N);
}